// LSTM_Layer_64287070487304
// MI455X (gfx1250) — compile-verified
//
#include <hip/hip_runtime.h>
#include <math.h>

// LSTM layer for MI455X (gfx1250, wave32).
// B=64, T=512, D=1024, U=1024. fp32 throughout via V_WMMA_F32_16X16X4_F32
// (exact fp32 matrix math, matches the fp32 reference; Wx/Wh = 32MB stay in
// the 192MB L2 so the recurrence is compute/latency bound, not HBM bound).
//
// One kernel launch per timestep (device-wide sync between steps).
// Each step computes z = x_t @ Wx + h_prev @ Wh + b fused, then the gate
// nonlinearities and the c/h update. h history lives directly in d_out;
// c ping-pongs between two [B,U] buffers in d_ws.
//
// Data movement: weights are constant across all 512 steps, so they are
// repacked ONCE per call into a WMMA-fragment-native layout
//   P[kk/2][col][gate][2]  (8 contiguous floats per lane per K-pair)
// letting the hot loop fetch all 4 gate B-fragments with two b128 loads
// (0.25 VMEM instructions per WMMA). Falls back to direct row-major loads
// if ws_size is too small for the packed copies.

typedef float v2f __attribute__((ext_vector_type(2)));
typedef float v4f __attribute__((ext_vector_type(4)));
typedef float v8f __attribute__((ext_vector_type(8)));

#define BB   64
#define TT   512
#define DD   1024
#define UU   1024
#define G4   (4 * UU)

__device__ __forceinline__ float sigmoid_f(float x) {
    return 1.0f / (1.0f + __expf(-x));
}

__device__ __forceinline__ v8f wmma_f32(v2f a, v2f b, v8f c) {
    return __builtin_amdgcn_wmma_f32_16x16x4_f32(false, a, false, b, (short)0,
                                                 c, false, false);
}

// ---------------------------------------------------------------------------
// One-time weight repack: W [K,4U] row-major -> P[kk2][col][g][2] with
// P[((kk2*U + col)*4 + g)*2 + j] = W[2*kk2 + j][g*U + col].
// ---------------------------------------------------------------------------
__global__ __launch_bounds__(256) void pack_weights_kernel(
    const float* __restrict__ W, float* __restrict__ P, int K)
{
    const size_t total = (size_t)(K / 2) * UU * 4;   // (kk2, col, g) triples
    size_t idx = (size_t)blockIdx.x * blockDim.x + threadIdx.x;
    if (idx >= total) return;
    const int    g   = (int)(idx & 3);
    const size_t tmp = idx >> 2;
    const int    col = (int)(tmp % UU);
    const size_t kk2 = tmp / UU;
    const float v0 = W[(2 * kk2)     * (size_t)G4 + (size_t)g * UU + col];
    const float v1 = W[(2 * kk2 + 1) * (size_t)G4 + (size_t)g * UU + col];
    P[idx * 2 + 0] = v0;
    P[idx * 2 + 1] = v1;
}

// ---------------------------------------------------------------------------
// Per-timestep kernel. Block: 64 threads = 2 waves; wave w owns rows
// [32w, 32w+32) as two 16-row tiles. blockIdx.x selects a 16-unit tile.
// Each wave holds 8 accumulators: acc[rowtile][gate] (16x16 f32 each).
// PACKED selects the fragment-native weight layout vs direct row-major.
// ---------------------------------------------------------------------------
template <bool PACKED>
__global__ __launch_bounds__(64) void lstm_step_kernel(
    const float* __restrict__ x,     // [B,T,D]
    const float* __restrict__ Wxp,   // packed or row-major Wx
    const float* __restrict__ Whp,   // packed or row-major Wh
    const float* __restrict__ bias,  // [4U]
    const float* __restrict__ hprev, // [B,*] row stride = hstride
    long long hstride,
    const float* __restrict__ cprev, // [B,U]
    float* __restrict__ cout,        // [B,U]
    float* __restrict__ out,         // [B,T,U]
    int t)
{
    const int lane = threadIdx.x & 31;
    const int wave = threadIdx.x >> 5;   // 0..1 -> 32-row slab
    const int u0   = blockIdx.x * 16;    // unit-tile base (16 units)
    const int row0 = wave * 32;
    const int m    = lane & 15;          // A-row / B-col / C-col within tile
    const int hi   = lane >> 4;          // selects K pair {0,1} vs {2,3}
    const int col  = u0 + m;

    // A-fragment source rows (per-lane): rows (row0+m) and (row0+16+m)
    const float* xrow0 = x + (size_t)(row0 + m) * (size_t)(TT * DD) + (size_t)t * DD;
    const float* xrow1 = xrow0 + (size_t)16 * (size_t)(TT * DD);
    const float* hrow0 = hprev + (size_t)(row0 + m) * (size_t)hstride;
    const float* hrow1 = hrow0 + (size_t)16 * (size_t)hstride;

    v8f acc[2][4];
#pragma unroll
    for (int rt = 0; rt < 2; ++rt)
#pragma unroll
        for (int g = 0; g < 4; ++g)
            acc[rt][g] = (v8f){};

    // ---- x_t @ Wx : K = DD ----
    for (int k = 0; k < DD; k += 4) {
        const int kk = k + 2 * hi;
        v2f a0, a1;
        a0.x = xrow0[kk]; a0.y = xrow0[kk + 1];
        a1.x = xrow1[kk]; a1.y = xrow1[kk + 1];

        v2f bm[4];
        if constexpr (PACKED) {
            const size_t kk2 = (size_t)(k / 2 + hi);
            const v4f* wp = (const v4f*)(Wxp + ((kk2 * UU + col) * 8));
            const v4f q0 = wp[0];           // gates 0,1 (j=0,1 each)
            const v4f q1 = wp[1];           // gates 2,3
            bm[0].x = q0.x; bm[0].y = q0.y;
            bm[1].x = q0.z; bm[1].y = q0.w;
            bm[2].x = q1.x; bm[2].y = q1.y;
            bm[3].x = q1.z; bm[3].y = q1.w;
        } else {
            const float* w0 = Wxp + (size_t)kk * G4 + col;
#pragma unroll
            for (int g = 0; g < 4; ++g) {
                bm[g].x = w0[g * UU];
                bm[g].y = w0[G4 + g * UU];
            }
        }
#pragma unroll
        for (int g = 0; g < 4; ++g) {
            acc[0][g] = wmma_f32(a0, bm[g], acc[0][g]);
            acc[1][g] = wmma_f32(a1, bm[g], acc[1][g]);
        }
    }

    // ---- h_{t-1} @ Wh : K = UU ----
    for (int k = 0; k < UU; k += 4) {
        const int kk = k + 2 * hi;
        v2f a0, a1;
        a0.x = hrow0[kk]; a0.y = hrow0[kk + 1];
        a1.x = hrow1[kk]; a1.y = hrow1[kk + 1];

        v2f bm[4];
        if constexpr (PACKED) {
            const size_t kk2 = (size_t)(k / 2 + hi);
            const v4f* wp = (const v4f*)(Whp + ((kk2 * UU + col) * 8));
            const v4f q0 = wp[0];
            const v4f q1 = wp[1];
            bm[0].x = q0.x; bm[0].y = q0.y;
            bm[1].x = q0.z; bm[1].y = q0.w;
            bm[2].x = q1.x; bm[2].y = q1.y;
            bm[3].x = q1.z; bm[3].y = q1.w;
        } else {
            const float* w0 = Whp + (size_t)kk * G4 + col;
#pragma unroll
            for (int g = 0; g < 4; ++g) {
                bm[g].x = w0[g * UU];
                bm[g].y = w0[G4 + g * UU];
            }
        }
#pragma unroll
        for (int g = 0; g < 4; ++g) {
            acc[0][g] = wmma_f32(a0, bm[g], acc[0][g]);
            acc[1][g] = wmma_f32(a1, bm[g], acc[1][g]);
        }
    }

    // ---- epilogue: gates, cell/hidden update ----
    float bv[4];
#pragma unroll
    for (int g = 0; g < 4; ++g)
        bv[g] = bias[g * UU + col];

#pragma unroll
    for (int rt = 0; rt < 2; ++rt) {
#pragma unroll
        for (int r = 0; r < 8; ++r) {
            const int row = row0 + rt * 16 + r + 8 * hi;  // C/D row M
            const float iv = sigmoid_f(acc[rt][0][r] + bv[0]);
            const float fv = sigmoid_f(acc[rt][1][r] + bv[1]);
            const float gv = tanhf(acc[rt][2][r] + bv[2]);
            const float ov = sigmoid_f(acc[rt][3][r] + bv[3]);

            const float cp = cprev[(size_t)row * UU + col];
            const float cn = fv * cp + iv * gv;
            const float hn = ov * tanhf(cn);

            cout[(size_t)row * UU + col] = cn;
            out[(size_t)row * (size_t)(TT * UU) + (size_t)t * UU + col] = hn;
        }
    }
}

extern "C" void kernel_launch(void* const* d_in, const int* in_sizes, int n_in,
                              void* d_out, int out_size, void* d_ws, size_t ws_size,
                              hipStream_t stream) {
    const float* x    = (const float*)d_in[0];  // [B,T,D]
    const float* h0   = (const float*)d_in[1];  // [B,U]
    const float* c0   = (const float*)d_in[2];  // [B,U]
    const float* Wx   = (const float*)d_in[3];  // [D,4U]
    const float* Wh   = (const float*)d_in[4];  // [U,4U]
    const float* bias = (const float*)d_in[5];  // [4U]
    float* out = (float*)d_out;                 // [B,T,U]

    // Workspace layout: [cbuf0 | cbuf1 | Wx packed | Wh packed]
    const size_t cElems    = (size_t)BB * UU;            // 64K floats
    const size_t packElems = (size_t)DD * G4;            // 4M floats (16MB) each
    const size_t needBytes = (2 * cElems + 2 * packElems) * sizeof(float);

    float* cbuf0 = (float*)d_ws;
    float* cbuf1 = cbuf0 + cElems;
    float* WxP   = cbuf1 + cElems;
    float* WhP   = WxP + packElems;

    const bool packed = (ws_size >= needBytes);

    if (packed) {
        const size_t total = (size_t)(DD / 2) * UU * 4;  // (kk2,col,gate)
        const int pb = 256;
        const int pg = (int)((total + pb - 1) / pb);
        pack_weights_kernel<<<pg, pb, 0, stream>>>(Wx, WxP, DD);
        pack_weights_kernel<<<pg, pb, 0, stream>>>(Wh, WhP, UU);
    }

    dim3 grid(UU / 16);   // 64 blocks
    dim3 block(64);       // 2 waves

    for (int t = 0; t < TT; ++t) {
        const float* hprev;
        long long hstride;
        if (t == 0) {
            hprev = h0;
            hstride = UU;
        } else {
            hprev = out + (size_t)(t - 1) * UU;   // h history lives in d_out
            hstride = (long long)TT * UU;
        }
        const float* cprev = (t == 0) ? c0 : (((t - 1) & 1) ? cbuf1 : cbuf0);
        float* cnext = (t & 1) ? cbuf1 : cbuf0;

        if (packed) {
            lstm_step_kernel<true><<<grid, block, 0, stream>>>(
                x, WxP, WhP, bias, hprev, hstride, cprev, cnext, out, t);
        } else {
            lstm_step_kernel<false><<<grid, block, 0, stream>>>(
                x, Wx, Wh, bias, hprev, hstride, cprev, cnext, out, t);
        }
    }
}